// Affinity_1735166788677
// MI455X (gfx1250) — compile-verified
//
#include <hip/hip_runtime.h>

typedef __attribute__((ext_vector_type(2))) float v2f;
typedef __attribute__((ext_vector_type(8))) float v8f;
typedef int v4i __attribute__((vector_size(16)));   // matches builtin param type

#define C_DIM 256
#define H_DIM 512
#define N_DIM 512

// ---------------------------------------------------------------------------
// gfx1250 async memory->LDS path (ASYNCcnt-tracked DMA), guarded so the file
// still compiles if this toolchain does not declare the builtins.
// Probe round 2 showed the builtin takes (v4i*, v4i*, imm, imm).
// ---------------------------------------------------------------------------
#if defined(__HIP_DEVICE_COMPILE__) &&                                       \
    __has_builtin(__builtin_amdgcn_global_load_async_to_lds_b128) &&         \
    __has_builtin(__builtin_amdgcn_s_wait_asynccnt)
#define USE_ASYNC_LDS 1
#else
#define USE_ASYNC_LDS 0
#endif

__device__ __forceinline__ void cp16_g2l(const float* g, float* l)
{
#if USE_ASYNC_LDS
    // global_load_async_to_lds_b128 : 16B per lane, ASYNCcnt-tracked
    __builtin_amdgcn_global_load_async_to_lds_b128((v4i*)g, (v4i*)l, 0, 0);
#else
    const float4 v = *(const float4*)g;
    *(float4*)l = v;
#endif
}

__device__ __forceinline__ void wait_async_lds()
{
#if USE_ASYNC_LDS
    __builtin_amdgcn_s_wait_asynccnt(0);
#endif
}

// ---------------------------------------------------------------------------
// D[MxN] = A[MxK] * B[KxN], row-major. One 16x16 tile per wave via
// V_WMMA_F32_16X16X4_F32 (exact f32). Two independent accumulator chains
// (even/odd K steps) to keep two WMMAs in flight per wave.
// Fragment layout (ISA 7.12.2):
//   A 16x4 : VGPR0 = K=0 (lanes0-15)/K=2 (lanes16-31), VGPR1 = K=1/K=3
//   B 4x16 : VGPR0 = row K=0 / K=2 across lane halves, VGPR1 = K=1/K=3
//   C/D    : VGPR r = rows M=r / M=r+8 across lane halves, N = lane
// ---------------------------------------------------------------------------
__global__ void wmma_gemm_nn(const float* __restrict__ A, int lda,
                             const float* __restrict__ B, int ldb,
                             float* __restrict__ D, int ldd, int K)
{
    const int lane = threadIdx.x & 31;
    const int half = lane >> 4;
    const int l    = lane & 15;
    const int m0   = (blockIdx.y * blockDim.y + threadIdx.y) * 16;
    const int n0   = blockIdx.x * 16;

    v8f c0 = {}, c1 = {};
    for (int k0 = 0; k0 < K; k0 += 8) {
        {
            const int ka = k0 + 2 * half;
            const v2f a = *(const v2f*)&A[(m0 + l) * lda + ka];
            v2f b;
            b.x = B[(ka    ) * ldb + n0 + l];
            b.y = B[(ka + 1) * ldb + n0 + l];
            c0 = __builtin_amdgcn_wmma_f32_16x16x4_f32(false, a, false, b,
                                                       (short)0, c0, false, false);
        }
        {
            const int ka = k0 + 4 + 2 * half;
            const v2f a = *(const v2f*)&A[(m0 + l) * lda + ka];
            v2f b;
            b.x = B[(ka    ) * ldb + n0 + l];
            b.y = B[(ka + 1) * ldb + n0 + l];
            c1 = __builtin_amdgcn_wmma_f32_16x16x4_f32(false, a, false, b,
                                                       (short)0, c1, false, false);
        }
    }
    #pragma unroll
    for (int r = 0; r < 8; ++r)
        D[(m0 + r + 8 * half) * ldd + n0 + l] = c0[r] + c1[r];
}

// ---------------------------------------------------------------------------
// D[MxN] = A[MxK] * B[NxK]^T (+ bias[n] broadcast over rows, if bias != null)
// ---------------------------------------------------------------------------
__global__ void wmma_gemm_nt_bias(const float* __restrict__ A, int lda,
                                  const float* __restrict__ B, int ldb,
                                  const float* __restrict__ bias,
                                  float* __restrict__ D, int ldd, int K)
{
    const int lane = threadIdx.x & 31;
    const int half = lane >> 4;
    const int l    = lane & 15;
    const int m0   = (blockIdx.y * blockDim.y + threadIdx.y) * 16;
    const int n0   = blockIdx.x * 16;

    v8f c0 = {}, c1 = {};
    for (int k0 = 0; k0 < K; k0 += 8) {
        {
            const int ka = k0 + 2 * half;
            const v2f a = *(const v2f*)&A[(m0 + l) * lda + ka];
            const v2f b = *(const v2f*)&B[(n0 + l) * ldb + ka];  // B^T
            c0 = __builtin_amdgcn_wmma_f32_16x16x4_f32(false, a, false, b,
                                                       (short)0, c0, false, false);
        }
        {
            const int ka = k0 + 4 + 2 * half;
            const v2f a = *(const v2f*)&A[(m0 + l) * lda + ka];
            const v2f b = *(const v2f*)&B[(n0 + l) * ldb + ka];
            c1 = __builtin_amdgcn_wmma_f32_16x16x4_f32(false, a, false, b,
                                                       (short)0, c1, false, false);
        }
    }
    const float bv = bias ? bias[n0 + l] : 0.0f;
    #pragma unroll
    for (int r = 0; r < 8; ++r)
        D[(m0 + r + 8 * half) * ldd + n0 + l] = c0[r] + c1[r] + bv;
}

// ---------------------------------------------------------------------------
// M[i,j] = b2 + sum_h relu(hxb[i,h] + hy[j,h]) * w2[h]
// 32x32 output tile per block (256 threads, 4 outputs/thread).
// Double-buffered h-chunks of 128, staged with async memory->LDS DMA and
// s_wait_asynccnt handoff (prefetch chunk c+1 while computing chunk c).
// lx rows: stride 128 (reads are lane-uniform broadcasts, no padding needed).
// ly rows: stride 132 floats = 528 B (16B-aligned for b128, 2-way worst-case
// bank conflict on the lane-indexed read).
// ---------------------------------------------------------------------------
#define TI 32
#define HH 128
#define LYP 132
#define NCHUNK (H_DIM / HH)

__global__ void bilinear_relu_reduce(const float* __restrict__ hxb,
                                     const float* __restrict__ hy,
                                     const float* __restrict__ w2,
                                     const float* __restrict__ b2,
                                     float* __restrict__ Mout)
{
    __shared__ float lx[2][TI][HH];
    __shared__ float ly[2][TI][LYP];
    __shared__ float sw2[H_DIM];

    const int t  = threadIdx.x;     // 0..255
    const int j  = t & 31;          // output column within tile (== lane)
    const int ib = t >> 5;          // 0..7 output row group
    const int i0 = blockIdx.y * TI;
    const int j0 = blockIdx.x * TI;

    // Stage all of w2 once (2 KB).
    sw2[t]       = w2[t];
    sw2[t + 256] = w2[t + 256];

    // Stage one h-chunk (32 rows x 128 cols of hxb and hy) into buffer `buf`.
    auto stage = [&](int chunk, int buf) {
        const int h0 = chunk * HH;
        #pragma unroll
        for (int q = 0; q < 4; ++q) {
            const int idx = t + 256 * q;          // 1024 x b128 per matrix
            const int row = idx >> 5;             // 0..31
            const int col = (idx & 31) * 4;       // 0..124, 16B-aligned
            cp16_g2l(&hxb[(size_t)(i0 + row) * H_DIM + h0 + col], &lx[buf][row][col]);
            cp16_g2l(&hy [(size_t)(j0 + row) * H_DIM + h0 + col], &ly[buf][row][col]);
        }
    };

    stage(0, 0);

    float acc0 = 0.f, acc1 = 0.f, acc2 = 0.f, acc3 = 0.f;

    for (int c = 0; c < NCHUNK; ++c) {
        wait_async_lds();       // my chunk-c DMAs have landed
        __syncthreads();        // everyone's have landed; prev compute done
        if (c + 1 < NCHUNK)
            stage(c + 1, (c + 1) & 1);   // prefetch into the other buffer

        const int    cb  = c & 1;
        const float* swc = &sw2[c * HH];
        #pragma unroll 4
        for (int h = 0; h < HH; ++h) {
            const float yv = ly[cb][j][h];
            const float wv = swc[h];
            const float v0 = fmaxf(lx[cb][ib     ][h] + yv, 0.f);
            const float v1 = fmaxf(lx[cb][ib +  8][h] + yv, 0.f);
            const float v2 = fmaxf(lx[cb][ib + 16][h] + yv, 0.f);
            const float v3 = fmaxf(lx[cb][ib + 24][h] + yv, 0.f);
            acc0 = fmaf(v0, wv, acc0);
            acc1 = fmaf(v1, wv, acc1);
            acc2 = fmaf(v2, wv, acc2);
            acc3 = fmaf(v3, wv, acc3);
        }
    }

    const float bb = b2[0];
    Mout[(size_t)(i0 + ib     ) * N_DIM + j0 + j] = acc0 + bb;
    Mout[(size_t)(i0 + ib +  8) * N_DIM + j0 + j] = acc1 + bb;
    Mout[(size_t)(i0 + ib + 16) * N_DIM + j0 + j] = acc2 + bb;
    Mout[(size_t)(i0 + ib + 24) * N_DIM + j0 + j] = acc3 + bb;
}

extern "C" void kernel_launch(void* const* d_in, const int* in_sizes, int n_in,
                              void* d_out, int out_size, void* d_ws, size_t ws_size,
                              hipStream_t stream)
{
    (void)in_sizes; (void)n_in; (void)out_size; (void)ws_size;

    const float* X   = (const float*)d_in[0];   // (512, 256)
    const float* Y   = (const float*)d_in[1];   // (512, 256)
    const float* Wsr = (const float*)d_in[2];   // (256, 256)
    const float* Wtg = (const float*)d_in[3];   // (256, 256)
    const float* W1  = (const float*)d_in[4];   // (512, 512)
    const float* b1  = (const float*)d_in[5];   // (512,)
    const float* w2  = (const float*)d_in[6];   // (1, 512)
    const float* b2  = (const float*)d_in[7];   // (1,)
    float* Mout = (float*)d_out;                // (512, 512)

    // Workspace: Gx(512x256) Gy(512x256) hxb(512x512) hy(512x512) = 3 MB
    float* Gx  = (float*)d_ws;
    float* Gy  = Gx  + (size_t)H_DIM * C_DIM;
    float* hxb = Gy  + (size_t)H_DIM * C_DIM;
    float* hy  = hxb + (size_t)N_DIM * H_DIM;

    const dim3 blk(32, 4);   // 4 waves, one 16x16 tile each

    // Gx = W1[:, :C] @ W_sr ; Gy = W1[:, C:] @ W_tg   (512x256, K=256)
    wmma_gemm_nn<<<dim3(C_DIM / 16, H_DIM / 64), blk, 0, stream>>>(
        W1,         2 * C_DIM, Wsr, C_DIM, Gx, C_DIM, C_DIM);
    wmma_gemm_nn<<<dim3(C_DIM / 16, H_DIM / 64), blk, 0, stream>>>(
        W1 + C_DIM, 2 * C_DIM, Wtg, C_DIM, Gy, C_DIM, C_DIM);

    // hxb = X @ Gx^T + b1 ; hy = Y @ Gy^T   (512x512, K=256)
    wmma_gemm_nt_bias<<<dim3(H_DIM / 16, N_DIM / 64), blk, 0, stream>>>(
        X, C_DIM, Gx, C_DIM, b1,      hxb, H_DIM, C_DIM);
    wmma_gemm_nt_bias<<<dim3(H_DIM / 16, N_DIM / 64), blk, 0, stream>>>(
        Y, C_DIM, Gy, C_DIM, nullptr, hy,  H_DIM, C_DIM);

    // M[i,j] = b2 + sum_h relu(hxb[i,h] + hy[j,h]) * w2[h]
    bilinear_relu_reduce<<<dim3(N_DIM / TI, N_DIM / TI), 256, 0, stream>>>(
        hxb, hy, w2, b2, Mout);
}